// WHNN_aggregation_layer_63290638074193
// MI455X (gfx1250) — compile-verified
//
#include <hip/hip_runtime.h>
#include <hip/hip_bf16.h>

typedef __attribute__((ext_vector_type(2))) float v2f;
typedef __attribute__((ext_vector_type(8))) float v8f;

#define NND   16384   // nodes == edges
#define PP    32      // projections
#define CC    64      // channels
#define LL    65536   // incidences (N*K)
#define MM    128     // anchors

// ---- workspace layout (float offsets) ----
#define OFF_WHATA   0u          // 32*64
#define OFF_WHATB   2048u       // 32*64
#define OFF_BV2E    4096u       // 32*4
#define OFF_CV2E    4224u       // 32
#define OFF_BE2V    4256u       // 32*4
#define OFF_CE2V    4384u       // 32
#define OFF_WFOLD   4416u       // 64*32 -> 6464
#define OFF_Y       8192u       // N*32
#define OFF_EPOOL   532480u     // N*32
#define OFF_Z       1056768u    // N*32
#define OFF_VPOOL   1581056u    // N*32
#define OFF_H       2105344u    // 64*N
#define OFF_STATS   3153920u    // 64*2
#define OFF_INT     3154048u    // int region: cntE[N], cntV[N], memE[4N], memV[4N]

// -------------------------------------------------------------------------
// Prep: normalized projection rows, closed-form interp constants, folded MLP
// -------------------------------------------------------------------------
__global__ void prep_kernel(const float* __restrict__ thetaA, const float* __restrict__ wA,
                            const float* __restrict__ thetaB, const float* __restrict__ wB,
                            const float* __restrict__ mlp_w,  const float* __restrict__ vproj,
                            float* __restrict__ ws) {
  int t = threadIdx.x;
  float* whatA = ws + OFF_WHATA;
  float* whatB = ws + OFF_WHATB;
  if (t < 64) {
    int p = t & 31;
    const float* th = (t < 32) ? thetaA : thetaB;
    float*       wh = (t < 32) ? whatA  : whatB;
    float s = 0.f;
    for (int c = 0; c < CC; ++c) { float v = th[p*CC+c]; s += v*v; }
    float inv = rsqrtf(s);
    for (int c = 0; c < CC; ++c) wh[p*CC+c] = th[p*CC+c]*inv;

    // fixed interpolation weights (D==4 for every group)
    const float* w  = (t < 32) ? wA : wB;
    float* Bo = ws + ((t < 32) ? OFF_BV2E : OFF_BE2V);
    float* co = ws + ((t < 32) ? OFF_CV2E : OFF_CE2V);
    float b0=0.f,b1=0.f,b2=0.f,b3=0.f,cc=0.f;
    const float x1 = (1.0f/3.0f)*0.99999f + 1e-5f;
    const float x2 = (2.0f/3.0f)*0.99999f + 1e-5f;
    for (int m = 0; m < MM; ++m) {
      float q = ((float)m/127.0f)*0.99998f + 1e-5f;
      int hi = 1;
      if (x1 <= q) hi = 2;
      if (x2 <= q) hi = 3;
      float xlo = ((float)(hi-1)/3.0f)*0.99999f + 1e-5f;
      float xhi = ((float)hi   /3.0f)*0.99999f + 1e-5f;
      float tt  = (q - xlo)/(xhi - xlo);
      float wm  = w[p*MM+m];
      float alo = wm*(1.0f-tt), ahi = wm*tt;
      if      (hi==1) { b0 += alo; b1 += ahi; }
      else if (hi==2) { b1 += alo; b2 += ahi; }
      else            { b2 += alo; b3 += ahi; }
      cc += wm*(-1.0f + 2.0f*(float)m/127.0f);
    }
    const float invM = 1.0f/(float)MM;
    Bo[p*4+0]=b0*invM; Bo[p*4+1]=b1*invM; Bo[p*4+2]=b2*invM; Bo[p*4+3]=b3*invM;
    co[p]=cc*invM;
  }
  // Wfold[o,p] = sum_c mlp_w[o, C+c] * vertex_proj_w[c, p]
  float* Wfold = ws + OFF_WFOLD;
  for (int idx = t; idx < CC*PP; idx += blockDim.x) {
    int o = idx / PP, p = idx % PP;
    float s = 0.f;
    for (int c = 0; c < CC; ++c) s += mlp_w[o*(2*CC) + CC + c] * vproj[c*PP + p];
    Wfold[idx] = s;
  }
}

// -------------------------------------------------------------------------
// Build per-group member lists (slot order irrelevant: values get sorted)
// -------------------------------------------------------------------------
__global__ void build_kernel(const int* __restrict__ v_idx, const int* __restrict__ e_idx,
                             int* __restrict__ cntE, int* __restrict__ cntV,
                             int* __restrict__ memE, int* __restrict__ memV) {
  int l = blockIdx.x * blockDim.x + threadIdx.x;
  if (l >= LL) return;
  int e = e_idx[l], v = v_idx[l];
  int se = atomicAdd(&cntE[e], 1) & 3;
  memE[e*4 + se] = v;
  int sv = atomicAdd(&cntV[v], 1) & 3;
  memV[v*4 + sv] = e;
}

// -------------------------------------------------------------------------
// GEMM: out[row, p] = sum_c A_cm[c*NR+row] * W[p*C + c]   (f32 WMMA 16x16x4)
// A column-major-by-feature [C][NR], W row-major [32][64], out [NR][32]
// -------------------------------------------------------------------------
__global__ void gemm_proj_kernel(const float* __restrict__ A_cm,
                                 const float* __restrict__ W,
                                 float* __restrict__ out, int NR) {
  int lane = threadIdx.x & 31;
  int wv   = threadIdx.x >> 5;
  int tile = blockIdx.x * 4 + wv;
  int colTile = tile & 1;
  int rowTile = tile >> 1;
  int row0 = rowTile * 16, col0 = colTile * 16;
  int half = lane >> 4, mr = lane & 15;
  v8f acc = {0.f,0.f,0.f,0.f,0.f,0.f,0.f,0.f};
  #pragma unroll
  for (int kk = 0; kk < CC/4; ++kk) {
    int k0 = kk*4 + half*2;
    v2f a, b;
    a.x = A_cm[k0*NR + row0 + mr];
    a.y = A_cm[(k0+1)*NR + row0 + mr];
    b.x = W[(col0+mr)*CC + k0];
    b.y = W[(col0+mr)*CC + k0 + 1];
    acc = __builtin_amdgcn_wmma_f32_16x16x4_f32(false, a, false, b, (short)0, acc, false, false);
  }
  #pragma unroll
  for (int r = 0; r < 8; ++r)
    out[(row0 + r + half*8)*PP + col0 + mr] = acc[r];
}

// -------------------------------------------------------------------------
// Pool: 4-member gather, sort network, fixed 4-tap weighted sum
// -------------------------------------------------------------------------
__global__ void pool4_kernel(const float* __restrict__ proj,   // [N][32]
                             const int*   __restrict__ members,// [N][4]
                             const float* __restrict__ Bw,     // [32][4]
                             const float* __restrict__ cw,     // [32]
                             float* __restrict__ out) {        // [N][32]
  int idx = blockIdx.x * blockDim.x + threadIdx.x;
  if (idx >= NND*PP) return;
  int p = idx & 31;
  int g = idx >> 5;
  int m0 = members[g*4+0], m1 = members[g*4+1], m2 = members[g*4+2], m3 = members[g*4+3];
  float s0 = proj[m0*PP+p], s1 = proj[m1*PP+p], s2 = proj[m2*PP+p], s3 = proj[m3*PP+p];
  float t;
  if (s0 > s1) { t=s0; s0=s1; s1=t; }
  if (s2 > s3) { t=s2; s2=s3; s3=t; }
  if (s0 > s2) { t=s0; s0=s2; s2=t; }
  if (s1 > s3) { t=s1; s1=s3; s3=t; }
  if (s1 > s2) { t=s1; s1=s2; s2=t; }
  out[idx] = cw[p] - (Bw[p*4+0]*s0 + Bw[p*4+1]*s1 + Bw[p*4+2]*s2 + Bw[p*4+3]*s3);
}

// -------------------------------------------------------------------------
// GEMM: out_cm[c*NR + e] = sum_p pool[e*32+p] * Wp[c*32+p]  (f32 WMMA)
// -------------------------------------------------------------------------
__global__ void gemm_pool_to_cm(const float* __restrict__ pool,
                                const float* __restrict__ Wp,
                                float* __restrict__ out_cm, int NR) {
  int lane = threadIdx.x & 31;
  int wv   = threadIdx.x >> 5;
  int tile = blockIdx.x * 4 + wv;
  int colTile = tile & 3;           // C/16 = 4
  int rowTile = tile >> 2;
  int row0 = rowTile*16, col0 = colTile*16;
  int half = lane >> 4, mr = lane & 15;
  v8f acc = {0.f,0.f,0.f,0.f,0.f,0.f,0.f,0.f};
  #pragma unroll
  for (int kk = 0; kk < PP/4; ++kk) {
    int k0 = kk*4 + half*2;
    v2f a, b;
    a.x = pool[(row0+mr)*PP + k0];
    a.y = pool[(row0+mr)*PP + k0 + 1];
    b.x = Wp[(col0+mr)*PP + k0];
    b.y = Wp[(col0+mr)*PP + k0 + 1];
    acc = __builtin_amdgcn_wmma_f32_16x16x4_f32(false, a, false, b, (short)0, acc, false, false);
  }
  #pragma unroll
  for (int r = 0; r < 8; ++r)
    out_cm[(col0+mr)*NR + row0 + r + half*8] = acc[r];
}

// -------------------------------------------------------------------------
// MLP GEMM fused over K=96: h[o*N+n] = b[o] + W1.vf + Wfold.vert_pool
// -------------------------------------------------------------------------
__global__ void mlp_gemm_kernel(const float* __restrict__ vf_cm,  // [64][N]
                                const float* __restrict__ vpool,  // [N][32]
                                const float* __restrict__ W1,     // mlp_w [64][128]
                                const float* __restrict__ Wfold,  // [64][32]
                                const float* __restrict__ bias,   // [64]
                                float* __restrict__ h, int NR) {
  int lane = threadIdx.x & 31;
  int wv   = threadIdx.x >> 5;
  int tile = blockIdx.x * 4 + wv;
  int colTile = tile & 3;           // 64/16 = 4
  int rowTile = tile >> 2;
  int row0 = rowTile*16, col0 = colTile*16;
  int half = lane >> 4, mr = lane & 15;
  v8f acc = {0.f,0.f,0.f,0.f,0.f,0.f,0.f,0.f};
  #pragma unroll
  for (int kk = 0; kk < 24; ++kk) {   // K = 64 + 32 = 96
    int k0 = kk*4 + half*2;
    v2f a, b;
    if (k0 < 64) {
      a.x = vf_cm[k0*NR + row0 + mr];
      a.y = vf_cm[(k0+1)*NR + row0 + mr];
      b.x = W1[(col0+mr)*128 + k0];
      b.y = W1[(col0+mr)*128 + k0 + 1];
    } else {
      int kp = k0 - 64;
      a.x = vpool[(row0+mr)*PP + kp];
      a.y = vpool[(row0+mr)*PP + kp + 1];
      b.x = Wfold[(col0+mr)*PP + kp];
      b.y = Wfold[(col0+mr)*PP + kp + 1];
    }
    acc = __builtin_amdgcn_wmma_f32_16x16x4_f32(false, a, false, b, (short)0, acc, false, false);
  }
  float bv = bias[col0+mr];
  #pragma unroll
  for (int r = 0; r < 8; ++r)
    h[(col0+mr)*NR + row0 + r + half*8] = acc[r] + bv;
}

// -------------------------------------------------------------------------
// BatchNorm (batch stats) + LeakyReLU(0.2)
// -------------------------------------------------------------------------
__global__ void bn_reduce_kernel(const float* __restrict__ h, float* __restrict__ stats, int NR) {
  __shared__ float ssum[256], ssq[256];
  int o = blockIdx.x;
  float s = 0.f, q = 0.f;
  for (int n = threadIdx.x; n < NR; n += blockDim.x) {
    float v = h[o*NR + n]; s += v; q += v*v;
  }
  ssum[threadIdx.x] = s; ssq[threadIdx.x] = q;
  __syncthreads();
  for (int st = 128; st > 0; st >>= 1) {
    if ((int)threadIdx.x < st) {
      ssum[threadIdx.x] += ssum[threadIdx.x + st];
      ssq[threadIdx.x]  += ssq[threadIdx.x + st];
    }
    __syncthreads();
  }
  if (threadIdx.x == 0) {
    float mean = ssum[0] / (float)NR;
    float var  = ssq[0] / (float)NR - mean*mean;
    stats[o*2+0] = mean;
    stats[o*2+1] = rsqrtf(var + 1e-5f);
  }
}

__global__ void bn_apply_kernel(const float* __restrict__ h, const float* __restrict__ stats,
                                const float* __restrict__ gamma, const float* __restrict__ beta,
                                float* __restrict__ out, int NR) {
  int idx = blockIdx.x * blockDim.x + threadIdx.x;
  if (idx >= CC*NR) return;
  int o = idx >> 14;   // NR = 16384
  float v = (h[idx] - stats[o*2]) * stats[o*2+1] * gamma[o] + beta[o];
  out[idx] = (v >= 0.f) ? v : 0.2f*v;
}

// -------------------------------------------------------------------------
extern "C" void kernel_launch(void* const* d_in, const int* in_sizes, int n_in,
                              void* d_out, int out_size, void* d_ws, size_t ws_size,
                              hipStream_t stream) {
  const float* vertex_feat = (const float*)d_in[0];
  const int*   v_idx       = (const int*)  d_in[1];
  const int*   e_idx       = (const int*)  d_in[2];
  const float* theta_v2e   = (const float*)d_in[3];
  const float* w_v2e       = (const float*)d_in[4];
  const float* theta_e2v   = (const float*)d_in[5];
  const float* w_e2v       = (const float*)d_in[6];
  const float* edge_proj_w = (const float*)d_in[7];
  const float* vertex_proj = (const float*)d_in[8];
  const float* mlp_w       = (const float*)d_in[9];
  const float* mlp_b       = (const float*)d_in[10];
  const float* bn_gamma    = (const float*)d_in[11];
  const float* bn_beta     = (const float*)d_in[12];
  (void)in_sizes; (void)n_in; (void)out_size; (void)ws_size;

  float* ws = (float*)d_ws;
  float* out = (float*)d_out;
  float* edge_out = out + CC*NND;   // second output: edge_feat_out [C][N]

  int* ip   = (int*)(ws + OFF_INT);
  int* cntE = ip;
  int* cntV = ip + NND;
  int* memE = ip + 2*NND;
  int* memV = ip + 2*NND + 4*NND;

  // zero atomic counters (graph-capture safe)
  hipMemsetAsync(cntE, 0, 2*NND*sizeof(int), stream);

  prep_kernel<<<1, 256, 0, stream>>>(theta_v2e, w_v2e, theta_e2v, w_e2v, mlp_w, vertex_proj, ws);
  build_kernel<<<LL/256, 256, 0, stream>>>(v_idx, e_idx, cntE, cntV, memE, memV);

  // Y = vf . what_v2e^T      (2048 tiles, 4 waves/block)
  gemm_proj_kernel<<<512, 128, 0, stream>>>(vertex_feat, ws + OFF_WHATA, ws + OFF_Y, NND);
  pool4_kernel<<<(NND*PP)/256, 256, 0, stream>>>(ws + OFF_Y, memE, ws + OFF_BV2E, ws + OFF_CV2E, ws + OFF_EPOOL);

  // edge_feat_out = edge_pool . edge_proj_w^T  (written column-major into d_out)
  gemm_pool_to_cm<<<1024, 128, 0, stream>>>(ws + OFF_EPOOL, edge_proj_w, edge_out, NND);

  // Z = ef . what_e2v^T   (reads edge_feat_out column-major directly)
  gemm_proj_kernel<<<512, 128, 0, stream>>>(edge_out, ws + OFF_WHATB, ws + OFF_Z, NND);
  pool4_kernel<<<(NND*PP)/256, 256, 0, stream>>>(ws + OFF_Z, memV, ws + OFF_BE2V, ws + OFF_CE2V, ws + OFF_VPOOL);

  // h = mlp_w[:, :64].vf + Wfold.vert_pool + b
  mlp_gemm_kernel<<<1024, 128, 0, stream>>>(vertex_feat, ws + OFF_VPOOL, mlp_w, ws + OFF_WFOLD,
                                            mlp_b, ws + OFF_H, NND);
  bn_reduce_kernel<<<CC, 256, 0, stream>>>(ws + OFF_H, ws + OFF_STATS, NND);
  bn_apply_kernel<<<(CC*NND)/256, 256, 0, stream>>>(ws + OFF_H, ws + OFF_STATS, bn_gamma, bn_beta, out, NND);
}